// ProposalLayer_17446157157174
// MI455X (gfx1250) — compile-verified
//
#include <hip/hip_runtime.h>
#include <stdint.h>
#include <math.h>

// ---------------------------------------------------------------------------
// Problem constants (from reference)
// ---------------------------------------------------------------------------
#define RPN_B      16
#define RPN_H      64
#define RPN_W      64
#define RPN_A      9
#define RPN_HW     (RPN_H * RPN_W)                 // 4096
#define RPN_N      (RPN_HW * RPN_A)                // 36864 anchors / batch
#define PRE_TOPK   2000
#define TOPK_PAD   2048
#define POST_TOPK  300
#define NMS_THR    0.7f

// Base anchors for BASE_SIZE=16, ratios {0.5,1,2}, scales {1,2,4}
// (precomputed to match numpy's _generate_anchors exactly)
__constant__ float c_anchors[RPN_A * 4] = {
    -3.5f,   2.0f, 18.5f, 13.0f,   // r=0.5 s=1
   -15.0f,  -4.0f, 30.0f, 19.0f,   // r=0.5 s=2
   -38.0f, -16.0f, 53.0f, 31.0f,   // r=0.5 s=4
     0.0f,   0.0f, 15.0f, 15.0f,   // r=1   s=1
    -8.0f,  -8.0f, 23.0f, 23.0f,   // r=1   s=2
   -24.0f, -24.0f, 39.0f, 39.0f,   // r=1   s=4
     2.5f,  -3.0f, 12.5f, 18.0f,   // r=2   s=1
    -3.0f, -14.0f, 18.0f, 29.0f,   // r=2   s=2
   -14.0f, -36.0f, 29.0f, 51.0f,   // r=2   s=4
};

// ---------------------------------------------------------------------------
// CDNA5 feature detection (compile-safe fallbacks everywhere)
// ---------------------------------------------------------------------------
#ifdef __has_builtin
#if __has_builtin(__builtin_amdgcn_global_load_async_to_lds_b32) && \
    __has_builtin(__builtin_amdgcn_s_wait_asynccnt)
#define RPN_HAS_ASYNC 1
#endif
#if __has_builtin(__builtin_amdgcn_tensor_load_to_lds) && \
    __has_builtin(__builtin_amdgcn_s_wait_tensorcnt)
#define RPN_HAS_TDM 1
#endif
#endif
#ifndef RPN_HAS_ASYNC
#define RPN_HAS_ASYNC 0
#endif
#ifndef RPN_HAS_TDM
#define RPN_HAS_TDM 0
#endif

#if RPN_HAS_TDM
typedef unsigned int rpn_u32x4 __attribute__((ext_vector_type(4)));
typedef int          rpn_i32x4 __attribute__((ext_vector_type(4)));
typedef int          rpn_i32x8 __attribute__((ext_vector_type(8)));
#endif

// Async copy of one dword from global to LDS (ASYNCcnt path on gfx1250).
// Builtin params are addrspace-qualified int* (per hipcc diagnostic).
__device__ __forceinline__ void rpn_async_cp_b32(uint32_t* lds_dst,
                                                 const uint32_t* g_src) {
#if RPN_HAS_ASYNC
  __builtin_amdgcn_global_load_async_to_lds_b32(
      (__attribute__((address_space(1))) int*)(uintptr_t)g_src,
      (__attribute__((address_space(3))) int*)(
          __attribute__((address_space(3))) void*)lds_dst,
      /*offset=*/0, /*cpol=*/0);
#else
  *lds_dst = *g_src;
#endif
}

__device__ __forceinline__ void rpn_async_wait() {
#if RPN_HAS_ASYNC
  __builtin_amdgcn_s_wait_asynccnt(0);
#endif
}

// ---------------------------------------------------------------------------
// Box decode (matches reference _decode + clip)
// n = (y*W + x)*A + a
// ---------------------------------------------------------------------------
__device__ __forceinline__ void rpn_decode(const float* __restrict__ bbox, int b,
                                           int n, float& x1, float& y1,
                                           float& x2, float& y2) {
  int a    = n % RPN_A;
  int cell = n / RPN_A;
  int x    = cell & (RPN_W - 1);
  int y    = cell >> 6;

  float ax1 = c_anchors[a * 4 + 0] + (float)(x * 16);
  float ay1 = c_anchors[a * 4 + 1] + (float)(y * 16);
  float ax2 = c_anchors[a * 4 + 2] + (float)(x * 16);
  float ay2 = c_anchors[a * 4 + 3] + (float)(y * 16);
  float aw  = ax2 - ax1 + 1.0f;
  float ah  = ay2 - ay1 + 1.0f;
  float acx = ax1 + 0.5f * aw;
  float acy = ay1 + 0.5f * ah;

  const float* d =
      bbox + ((size_t)b * (4 * RPN_A) + 4 * a) * RPN_HW + (y * RPN_W + x);
  float d0 = d[0];
  float d1 = d[RPN_HW];
  float d2 = d[2 * RPN_HW];
  float d3 = d[3 * RPN_HW];

  float cx = d0 * aw + acx;
  float cy = d1 * ah + acy;
  float pw = expf(d2) * aw;
  float ph = expf(d3) * ah;

  x1 = fminf(fmaxf(cx - 0.5f * pw, 0.0f), 1023.0f);
  y1 = fminf(fmaxf(cy - 0.5f * ph, 0.0f), 1023.0f);
  x2 = fminf(fmaxf(cx + 0.5f * pw, 0.0f), 1023.0f);
  y2 = fminf(fmaxf(cy + 0.5f * ph, 0.0f), 1023.0f);
}

// ---------------------------------------------------------------------------
// K1: decode every anchor, build order-preserving sort key
// ---------------------------------------------------------------------------
__global__ void rpn_decode_keys_kernel(const float* __restrict__ labels,
                                       const float* __restrict__ bbox,
                                       uint32_t* __restrict__ keys) {
  int gid = blockIdx.x * 256 + threadIdx.x;
  if (gid >= RPN_B * RPN_N) return;
  int b = gid / RPN_N;
  int n = gid - b * RPN_N;

  int a    = n % RPN_A;
  int cell = n / RPN_A;
  int x    = cell & (RPN_W - 1);
  int y    = cell >> 6;

  float score =
      labels[((size_t)b * (2 * RPN_A) + (2 * a + 1)) * RPN_HW + (y * RPN_W + x)];

  float x1, y1, x2, y2;
  rpn_decode(bbox, b, n, x1, y1, x2, y2);
  bool keep = ((x2 - x1 + 1.0f) >= 16.0f) && ((y2 - y1 + 1.0f) >= 16.0f);
  float m = keep ? score : -__builtin_inff();

  uint32_t u = __float_as_uint(m);
  keys[gid] = (u & 0x80000000u) ? ~u : (u | 0x80000000u);
}

// ---------------------------------------------------------------------------
// K2: per-batch exact top-2000 via 4x8-bit MSB radix select (async-LDS staged
//     histograms), then LDS bitonic sort of 2048 (key,idx) descending.
// ---------------------------------------------------------------------------
__global__ __launch_bounds__(1024) void rpn_topk_kernel(
    const uint32_t* __restrict__ keys, uint32_t* __restrict__ tk_idx,
    float* __restrict__ tk_score) {
  const int b   = blockIdx.x;
  const int tid = threadIdx.x;

  __shared__ uint32_t hist[256];
  __shared__ uint32_t stage[4096];
  __shared__ uint32_t s_key[TOPK_PAD];
  __shared__ uint32_t s_idx[TOPK_PAD];
  __shared__ uint32_t sh_prefix, sh_need, sh_gt, sh_eq;

  const uint32_t* kb = keys + (size_t)b * RPN_N;

  if (tid == 0) { sh_prefix = 0u; sh_need = PRE_TOPK; }
  __syncthreads();

  // ---- radix select: find the exact 2000th-largest key -------------------
  for (int pass = 0; pass < 4; ++pass) {
    const int shift = 24 - 8 * pass;
    if (tid < 256) hist[tid] = 0u;
    __syncthreads();
    const uint32_t pref = sh_prefix;
    const uint32_t msk  = pass ? (0xFFFFFFFFu << (shift + 8)) : 0u;

    for (int base = 0; base < RPN_N; base += 4096) {
      for (int u = tid; u < 4096; u += 1024)
        rpn_async_cp_b32(&stage[u], kb + base + u);
      rpn_async_wait();
      __syncthreads();
      for (int u = tid; u < 4096; u += 1024) {
        uint32_t k = stage[u];
        if (((k ^ pref) & msk) == 0u)
          atomicAdd(&hist[(k >> shift) & 255u], 1u);
      }
      __syncthreads();
    }

    if (tid == 0) {
      uint32_t need = sh_need, cum = 0u;
      int d = 255;
      for (;;) {
        uint32_t c = hist[d];
        if (cum + c >= need || d == 0) break;
        cum += c;
        --d;
      }
      sh_need   = need - cum;
      sh_prefix = pref | ((uint32_t)d << shift);
    }
    __syncthreads();
  }

  const uint32_t T      = sh_prefix;      // exact threshold key
  const uint32_t needEq = sh_need;        // how many == T to take
  const uint32_t cntGt  = PRE_TOPK - needEq;

  if (tid == 0) { sh_gt = 0u; sh_eq = 0u; }
  for (int t = tid; t < TOPK_PAD; t += 1024) { s_key[t] = 0u; s_idx[t] = 0u; }
  __syncthreads();

  // ---- compaction: keys > T, plus needEq keys == T -----------------------
  for (int base = 0; base < RPN_N; base += 4096) {
    for (int u = tid; u < 4096; u += 1024)
      rpn_async_cp_b32(&stage[u], kb + base + u);
    rpn_async_wait();
    __syncthreads();
    for (int u = tid; u < 4096; u += 1024) {
      uint32_t k = stage[u];
      if (k > T) {
        uint32_t p = atomicAdd(&sh_gt, 1u);
        s_key[p]   = k;
        s_idx[p]   = (uint32_t)(base + u);
      } else if (k == T) {
        uint32_t p = atomicAdd(&sh_eq, 1u);
        if (p < needEq) {
          s_key[cntGt + p] = k;
          s_idx[cntGt + p] = (uint32_t)(base + u);
        }
      }
    }
    __syncthreads();
  }

  // ---- bitonic sort, 2048 elems, descending by key -----------------------
  for (int sz = 2; sz <= TOPK_PAD; sz <<= 1) {
    for (int j = sz >> 1; j > 0; j >>= 1) {
      __syncthreads();
      for (int t = tid; t < TOPK_PAD; t += 1024) {
        int ixj = t ^ j;
        if (ixj > t) {
          uint32_t ka = s_key[t], kb2 = s_key[ixj];
          bool up = ((t & sz) == 0);
          if (up ? (ka < kb2) : (ka > kb2)) {
            s_key[t]   = kb2;
            s_key[ixj] = ka;
            uint32_t ia = s_idx[t];
            s_idx[t]    = s_idx[ixj];
            s_idx[ixj]  = ia;
          }
        }
      }
    }
  }
  __syncthreads();

  for (int t = tid; t < TOPK_PAD; t += 1024) {
    uint32_t k = s_key[t];
    uint32_t u = (k & 0x80000000u) ? (k & 0x7FFFFFFFu) : ~k;
    tk_score[(size_t)b * TOPK_PAD + t] = __uint_as_float(u);
    tk_idx[(size_t)b * TOPK_PAD + t]   = s_idx[t];
  }
}

// ---------------------------------------------------------------------------
// K3: per-batch greedy NMS entirely in LDS; write 300 outputs.
//     Scores staged into LDS via a TDM tensor_load_to_lds descriptor.
// ---------------------------------------------------------------------------
__global__ __launch_bounds__(1024) void rpn_nms_kernel(
    const float* __restrict__ bbox, const uint32_t* __restrict__ tk_idx,
    const float* __restrict__ tk_score, float* __restrict__ out) {
  const int b   = blockIdx.x;
  const int tid = threadIdx.x;

  __shared__ float bx1[TOPK_PAD], by1[TOPK_PAD], bx2[TOPK_PAD], by2[TOPK_PAD];
  __shared__ float bar[TOPK_PAD], bsc[TOPK_PAD];
  __shared__ uint32_t rem[64];     // 2048-bit suppression bitmap
  __shared__ uint32_t wpref[64];   // per-word keep prefix counts
  __shared__ uint32_t s_total;

  // ---- stage scores: TDM DMA (TENSORcnt) or plain fallback ---------------
#if RPN_HAS_TDM
  if (tid < 32) {
    unsigned lds_off = (unsigned)(unsigned long long)(
        __attribute__((address_space(3))) float*)bsc;
    unsigned long long ga =
        (unsigned long long)(const void*)(tk_score + (size_t)b * TOPK_PAD);
    // D# group0: count=1, lds_addr, global_addr[56:0], type=2 (bits 127:126)
    rpn_u32x4 g0 = {1u, lds_off, (unsigned)(ga & 0xFFFFFFFFull),
                    (unsigned)((ga >> 32) & 0x01FFFFFFull) | 0x80000000u};
    // D# group1: data_size=4B; tensor_dim0=2048, tensor_dim1=1,
    //            tile_dim0=2048, tile_dim1=1, stride0=2048
    rpn_i32x8 g1 = {0x00020000,
                    (int)(2048u << 16),   // tensor_dim0[15:0] in [31:16]
                    (int)(1u << 16),      // tensor_dim1[15:0] in [31:16]
                    (int)(2048u << 16),   // tile_dim0 in [31:16]
                    1,                    // tile_dim1=1, tile_dim2=0
                    2048,                 // tensor_dim0_stride[31:0]
                    0, 0};
    rpn_i32x4 gz = {0, 0, 0, 0};
#if __clang_major__ >= 23
    rpn_i32x8 gz8 = {0, 0, 0, 0, 0, 0, 0, 0};
    __builtin_amdgcn_tensor_load_to_lds(g0, g1, gz, gz, gz8, 0);
#else
    __builtin_amdgcn_tensor_load_to_lds(g0, g1, gz, gz, 0);
#endif
    __builtin_amdgcn_s_wait_tensorcnt(0);
  }
#else
  for (int t = tid; t < TOPK_PAD; t += 1024)
    bsc[t] = tk_score[(size_t)b * TOPK_PAD + t];
#endif

  // ---- re-decode the selected 2000 boxes into LDS SoA --------------------
  for (int t = tid; t < TOPK_PAD; t += 1024) {
    if (t < PRE_TOPK) {
      uint32_t n = tk_idx[(size_t)b * TOPK_PAD + t];
      float x1, y1, x2, y2;
      rpn_decode(bbox, b, (int)n, x1, y1, x2, y2);
      bx1[t] = x1; by1[t] = y1; bx2[t] = x2; by2[t] = y2;
      bar[t] = (x2 - x1 + 1.0f) * (y2 - y1 + 1.0f);
    } else {
      bx1[t] = -1.0e30f; by1[t] = -1.0e30f;
      bx2[t] = -1.0e30f; by2[t] = -1.0e30f;
      bar[t] = 0.0f;
    }
  }
  if (tid < 64) rem[tid] = 0u;
  __syncthreads();

  // ---- greedy suppression (matches reference scan exactly) ---------------
  for (int i = 0; i < PRE_TOPK; ++i) {
    bool curKeep = ((rem[i >> 5] >> (i & 31)) & 1u) == 0u;
    if (curKeep) {
      float x1 = bx1[i], y1 = by1[i], x2 = bx2[i], y2 = by2[i], A0 = bar[i];
      for (int t = tid; t < PRE_TOPK; t += 1024) {
        if (t > i) {
          float iw = fminf(x2, bx2[t]) - fmaxf(x1, bx1[t]) + 1.0f;
          float ih = fminf(y2, by2[t]) - fmaxf(y1, by1[t]) + 1.0f;
          iw = fmaxf(iw, 0.0f);
          ih = fmaxf(ih, 0.0f);
          float inter = iw * ih;
          if (inter > NMS_THR * (A0 + bar[t] - inter))
            atomicOr(&rem[t >> 5], 1u << (t & 31));
        }
      }
    }
    __syncthreads();
  }

  // ---- rank kept boxes (score order == slot order) -----------------------
  if (tid == 0) {
    uint32_t acc = 0u;
    for (int w = 0; w < 63; ++w) {
      wpref[w]    = acc;
      uint32_t kw = ~rem[w];
      if (w == 62) kw &= 0xFFFFu;  // bits 1984..1999 only
      acc += (uint32_t)__popc(kw);
    }
    s_total = acc;
  }
  __syncthreads();
  const uint32_t total = s_total;

  float* obox = out;                                 // (B,300,4)
  float* oscr = out + (size_t)RPN_B * POST_TOPK * 4; // (B,300)

  for (int t = tid; t < PRE_TOPK; t += 1024) {
    uint32_t w = (uint32_t)(t >> 5), bit = (uint32_t)(t & 31);
    if (((rem[w] >> bit) & 1u) == 0u) {
      uint32_t rank = wpref[w] + (uint32_t)__popc((~rem[w]) & ((1u << bit) - 1u));
      if (rank < POST_TOPK) {
        size_t o  = ((size_t)b * POST_TOPK + rank) * 4;
        obox[o + 0] = bx1[t];
        obox[o + 1] = by1[t];
        obox[o + 2] = bx2[t];
        obox[o + 3] = by2[t];
        oscr[(size_t)b * POST_TOPK + rank] = bsc[t];
      }
    }
  }
  for (int s = tid; s < POST_TOPK; s += 1024) {
    if ((uint32_t)s >= total) {
      size_t o  = ((size_t)b * POST_TOPK + s) * 4;
      obox[o + 0] = 0.0f;
      obox[o + 1] = 0.0f;
      obox[o + 2] = 0.0f;
      obox[o + 3] = 0.0f;
      oscr[(size_t)b * POST_TOPK + s] = 0.0f;
    }
  }
}

// ---------------------------------------------------------------------------
// Host launcher
// ---------------------------------------------------------------------------
extern "C" void kernel_launch(void* const* d_in, const int* in_sizes, int n_in,
                              void* d_out, int out_size, void* d_ws,
                              size_t ws_size, hipStream_t stream) {
  (void)in_sizes; (void)n_in; (void)out_size; (void)ws_size;
  const float* labels = (const float*)d_in[0];  // (16, 18, 64, 64) f32
  const float* bbox   = (const float*)d_in[1];  // (16, 36, 64, 64) f32

  // workspace layout: keys | topk_idx | topk_scores  (~2.6 MB total)
  uint32_t* keys     = (uint32_t*)d_ws;
  uint32_t* tk_idx   = keys + (size_t)RPN_B * RPN_N;
  float*    tk_score = (float*)(tk_idx + (size_t)RPN_B * TOPK_PAD);

  int total = RPN_B * RPN_N;
  rpn_decode_keys_kernel<<<(total + 255) / 256, 256, 0, stream>>>(labels, bbox,
                                                                  keys);
  rpn_topk_kernel<<<RPN_B, 1024, 0, stream>>>(keys, tk_idx, tk_score);
  rpn_nms_kernel<<<RPN_B, 1024, 0, stream>>>(bbox, tk_idx, tk_score,
                                             (float*)d_out);
}